// avg_45286135169789
// MI455X (gfx1250) — compile-verified
//
#include <hip/hip_runtime.h>
#include <hip/hip_bf16.h>
#include <math.h>

#define N_NODES 50000
#define N_EDGES 800000
#define ICH 256
#define HID 128
#define OCH 64
#define NSTRIPS (N_NODES / 16)   // 3125 strips of 16 rows

typedef float v2f __attribute__((ext_vector_type(2)));
typedef float v8f __attribute__((ext_vector_type(8)));

// ---------------------------------------------------------------- init: deg=1, s=0
__global__ void init_k(float* __restrict__ deg, float* __restrict__ s) {
    int tid = blockIdx.x * blockDim.x + threadIdx.x;
    if (tid < N_NODES) deg[tid] = 1.0f;
    if (tid < HID) s[tid] = 0.0f;
}

// ---------------------------------------------------------------- deg[dst] += 1
__global__ void edge_deg_k(const int* __restrict__ ei, float* __restrict__ deg) {
    int e = blockIdx.x * blockDim.x + threadIdx.x;
    if (e < N_EDGES) atomicAdd(&deg[ei[N_EDGES + e]], 1.0f);
}

// ---------------------------------------------------------------- dis = rsqrt(deg); wsum = dis^2
__global__ void node_dis_k(float* __restrict__ dis, float* __restrict__ wsum) {
    int i = blockIdx.x * blockDim.x + threadIdx.x;
    if (i < N_NODES) {
        float d = rsqrtf(dis[i]);   // dis buffer currently holds deg
        dis[i]  = d;
        wsum[i] = d * d;
    }
}

// ---------------------------------------------------------------- xw = x @ W1  (fp32 WMMA)
__global__ __launch_bounds__(256) void gemm_xw_k(const float* __restrict__ x,
                                                 const float* __restrict__ W1,
                                                 float* __restrict__ xw) {
    // 64KB LDS: 128 k-rows x 128 n, permuted so (k, k+1) at fixed n are adjacent:
    //   idx(kk, n) = (kk>>1)*256 + n*2 + (kk&1)
    __shared__ float wlds[16384];

    const int tid  = threadIdx.x;
    const int wave = tid >> 5;
    const int lane = tid & 31;
    const int m    = lane & 15;     // column-within-tile / row-within-strip
    const int hh   = lane >> 4;     // half-wave selector (K pair)

    const int  strip = blockIdx.x * 8 + wave;
    const bool valid = strip < NSTRIPS;
    const int  arow  = valid ? strip * 16 + m : 0;
    const float* xrow = x + (size_t)arow * ICH;

    v8f c[8] = {};                  // 8 N-tiles of 16x16 f32 accum

    for (int st = 0; st < 2; ++st) {         // two 128-row K stages of W1
        __syncthreads();
#pragma unroll 4
        for (int i = 0; i < 64; ++i) {       // 16384 elems / 256 threads
            int flat = i * 256 + tid;
            int kk   = flat >> 7;
            int n    = flat & 127;
            wlds[(kk >> 1) * 256 + n * 2 + (kk & 1)] = W1[(size_t)(st * 128 + kk) * HID + n];
        }
        __syncthreads();

#pragma unroll 4
        for (int k4 = 0; k4 < 32; ++k4) {    // K step = 4
            const int kbase = k4 * 4;
            // A frag (16x4 f32): lane holds rows m, K = {2*hh, 2*hh+1}
            v2f a = *(const v2f*)(xrow + st * 128 + kbase + hh * 2);
            const float* bb = &wlds[((kbase >> 1) + hh) * 256 + m * 2];
#pragma unroll
            for (int nt = 0; nt < 8; ++nt) {
                // B frag (4x16 f32): lane holds N = nt*16+m, K = {2*hh, 2*hh+1}
                v2f b = *(const v2f*)(bb + nt * 32);
                c[nt] = __builtin_amdgcn_wmma_f32_16x16x4_f32(
                    false, a, false, b, (short)0, c[nt], false, false);
            }
        }
    }

    if (!valid) return;
    const int row0 = strip * 16;
#pragma unroll
    for (int nt = 0; nt < 8; ++nt)
#pragma unroll
        for (int r = 0; r < 8; ++r)
            xw[(size_t)(row0 + hh * 8 + r) * HID + nt * 16 + m] = c[nt][r];
}

// ---------------------------------------------------------------- acc = xw*dis^2 + b1
__global__ void acc_init_k(const float* __restrict__ xw, const float* __restrict__ dis,
                           const float* __restrict__ b1, float* __restrict__ acc) {
    int tid = blockIdx.x * blockDim.x + threadIdx.x;
    if (tid < N_NODES * HID) {
        int i = tid >> 7, ch = tid & 127;
        float d = dis[i];
        acc[tid] = xw[tid] * d * d + b1[ch];
    }
}

// ------------------------------------------------- acc[dst] += xw[src]*norm ; wsum[src] += norm
__global__ void edge_agg_k(const int* __restrict__ ei, const float* __restrict__ xw,
                           const float* __restrict__ dis, float* __restrict__ acc,
                           float* __restrict__ wsum) {
    int e    = (blockIdx.x * blockDim.x + threadIdx.x) >> 5;   // one wave per edge
    int lane = threadIdx.x & 31;
    if (e >= N_EDGES) return;
    int src = ei[e];
    int dst = ei[N_EDGES + e];
    float nrm = dis[src] * dis[dst];
    if (lane == 0) atomicAdd(&wsum[src], nrm);
    const float4 v = *(const float4*)(xw + (size_t)src * HID + lane * 4);
    float* a = acc + (size_t)dst * HID + lane * 4;
    atomicAdd(a + 0, v.x * nrm);
    atomicAdd(a + 1, v.y * nrm);
    atomicAdd(a + 2, v.z * nrm);
    atomicAdd(a + 3, v.w * nrm);
}

// ---------------------------------------------------------------- s[c] = sum_i wsum[i]*relu(acc[i][c])
__global__ void reduce_s_k(const float* __restrict__ acc, const float* __restrict__ wsum,
                           float* __restrict__ s) {
    __shared__ float sm[256];
    int tid  = threadIdx.x;
    int ch   = tid & 127;
    int half = tid >> 7;
    float local = 0.0f;
    for (int i = blockIdx.x * 2 + half; i < N_NODES; i += gridDim.x * 2) {
        float v = acc[(size_t)i * HID + ch];
        v = v > 0.0f ? v : 0.0f;
        local += wsum[i] * v;
    }
    sm[tid] = local;
    __syncthreads();
    if (tid < 128) atomicAdd(&s[tid], sm[tid] + sm[tid + 128]);
}

// ---------------------------------------------------------------- mls = [ (s/N)@Wmu+bmu , (s/N)@Wls+bls ]
__global__ void final_k(const float* __restrict__ s,
                        const float* __restrict__ Wmu, const float* __restrict__ bmu,
                        const float* __restrict__ Wls, const float* __restrict__ bls,
                        float* __restrict__ mls) {
    int oc = threadIdx.x;
    if (oc >= OCH) return;
    const float invN = 1.0f / (float)N_NODES;
    float mu = bmu[oc];
    float ls = bls[oc];
    for (int ch = 0; ch < HID; ++ch) {
        float sv = s[ch] * invN;
        mu += sv * Wmu[ch * OCH + oc];
        ls += sv * Wls[ch * OCH + oc];
    }
    mls[oc]       = mu;
    mls[OCH + oc] = ls;
}

// ---------------------------------------------------------------- broadcast to [N,64] x2
__global__ void bcast_k(const float* __restrict__ mls, float* __restrict__ out) {
    int tid = blockIdx.x * blockDim.x + threadIdx.x;
    if (tid < N_NODES * OCH) {
        int oc = tid & 63;
        out[tid]                 = mls[oc];
        out[N_NODES * OCH + tid] = mls[OCH + oc];
    }
}

extern "C" void kernel_launch(void* const* d_in, const int* in_sizes, int n_in,
                              void* d_out, int out_size, void* d_ws, size_t ws_size,
                              hipStream_t stream) {
    (void)in_sizes; (void)n_in; (void)out_size; (void)ws_size;
    const float* x   = (const float*)d_in[0];
    const int*   ei  = (const int*)  d_in[1];
    const float* W1  = (const float*)d_in[2];
    const float* b1  = (const float*)d_in[3];
    const float* Wmu = (const float*)d_in[4];
    const float* bmu = (const float*)d_in[5];
    const float* Wls = (const float*)d_in[6];
    const float* bls = (const float*)d_in[7];
    float* out = (float*)d_out;

    float* ws   = (float*)d_ws;
    float* xw   = ws;                    // 6,400,000 f32
    float* acc  = ws + 6400000;          // 6,400,000 f32
    float* dis  = ws + 12800000;         //    50,000 f32 (deg, then dis in-place)
    float* wsum = ws + 12850000;         //    50,000 f32
    float* s    = ws + 12900000;         //       128 f32
    float* mls  = ws + 12900128;         //       128 f32

    init_k     <<<196,    256, 0, stream>>>(dis, s);
    edge_deg_k <<<3125,   256, 0, stream>>>(ei, dis);
    node_dis_k <<<196,    256, 0, stream>>>(dis, wsum);
    gemm_xw_k  <<<391,    256, 0, stream>>>(x, W1, xw);
    acc_init_k <<<25000,  256, 0, stream>>>(xw, dis, b1, acc);
    edge_agg_k <<<100000, 256, 0, stream>>>(ei, xw, dis, acc, wsum);
    reduce_s_k <<<256,    256, 0, stream>>>(acc, wsum, s);
    final_k    <<<1,       64, 0, stream>>>(s, Wmu, bmu, Wls, bls, mls);
    bcast_k    <<<12500,  256, 0, stream>>>(mls, out);
}